// DeformSegmentationModule_80951543595780
// MI455X (gfx1250) — compile-verified
//
#include <hip/hip_runtime.h>
#include <stdint.h>

// ---------------------------------------------------------------------------
// Problem constants (match reference)
// ---------------------------------------------------------------------------
constexpr int   kC      = 16;
constexpr int   kH      = 1024;
constexpr int   kW      = 1024;
constexpr int   kHW     = kH * kW;
constexpr float kHoleP  = 0.05f;
constexpr int   kNProp  = 8;

// Tiling: 32x32 output tile, halo 8 (propagation radius) -> 48x48 region.
constexpr int   kTS     = 32;
constexpr int   kHalo   = 8;
constexpr int   kR      = kTS + 2 * kHalo;   // 48
constexpr int   kRR     = kR * kR;           // 2304
constexpr int   kThreads= 256;
constexpr int   kPxT    = kRR / kThreads;    // 9 region pixels / thread
constexpr int   kSH     = 512;               // downsampled grid extent
constexpr int   kSN     = 26;                // s_small patch extent (24 + 2 halo)

// ---------------------------------------------------------------------------
// CDNA5 async global->LDS path (gfx1250), guarded so compile always succeeds.
// ---------------------------------------------------------------------------
#if defined(__has_builtin)
#if __has_builtin(__builtin_amdgcn_global_load_async_to_lds_b32)
#define HAVE_ASYNC_LDS 1
#endif
#if __has_builtin(__builtin_amdgcn_s_wait_asynccnt)
#define HAVE_WAIT_ASYNC 1
#endif
#endif

typedef __attribute__((address_space(1))) int* gptr_i32;
typedef __attribute__((address_space(3))) int* lptr_i32;

__device__ __forceinline__ void async_g2l_b32(const float* g, float* l) {
#if defined(HAVE_ASYNC_LDS)
  // AS1 via inttoptr (global generic addr == AS1 addr); AS3 via low-32-bit
  // truncation of the generic shared address (== workgroup LDS byte offset).
  __builtin_amdgcn_global_load_async_to_lds_b32(
      (gptr_i32)(uintptr_t)g,
      (lptr_i32)(uint32_t)(uintptr_t)l,
      0, 0);
#else
  *l = *g;
#endif
}

__device__ __forceinline__ void wait_async_zero() {
#if defined(HAVE_ASYNC_LDS)
#if defined(HAVE_WAIT_ASYNC)
  __builtin_amdgcn_s_wait_asynccnt(0);
#else
  asm volatile("s_wait_asynccnt 0" ::: "memory");
#endif
#endif
}

// ---------------------------------------------------------------------------
// Kernel: one block = one (tile, channel). All 8 propagation iterations in LDS.
// ---------------------------------------------------------------------------
__global__ __launch_bounds__(kThreads)
void hole_fill_kernel(const float* __restrict__ x,
                      const float* __restrict__ hu,
                      float* __restrict__ out) {
  __shared__ float s_small[kSN * kSN];  // channel-union invalid count @ even coords
  __shared__ float v_lds[kRR];
  __shared__ float w_lds[kRR];
  __shared__ float x_lds[kRR];
  __shared__ float u_lds[kRR];

  const int t  = threadIdx.x;
  const int c  = blockIdx.z;
  const int i0 = blockIdx.y * kTS;
  const int j0 = blockIdx.x * kTS;
  // Downsampled-grid patch base: region pixel gi maps to small row gi>>1,
  // gi in [i0-8, i0+39] -> small rows [i0/2-4, i0/2+19]; conv needs +/-1.
  const int A0 = (i0 >> 1) - (kHalo >> 1);   // i0/2 - 4
  const int B0 = (j0 >> 1) - (kHalo >> 1);

  const float* xc = x  + (size_t)c * kHW;
  const float* uc = hu + (size_t)c * kHW;

  // ---- 1) Kick off async region staging (x, hole_u) into LDS --------------
#pragma unroll
  for (int k = 0; k < kPxT; ++k) {
    const int idx = t + k * kThreads;
    const int r  = idx / kR, cc = idx % kR;
    const int gi = i0 - kHalo + r, gj = j0 - kHalo + cc;
    if (gi >= 0 && gi < kH && gj >= 0 && gj < kW) {
      const size_t go = (size_t)gi * kW + gj;
      async_g2l_b32(xc + go, &x_lds[idx]);
      async_g2l_b32(uc + go, &u_lds[idx]);
    }
  }

  // ---- 2) Overlap: channel-union invalid count on even-coordinate grid ----
  // s_small(a,b) = sum_ch [hu(ch, 2a, 2b) < p]; zero outside the 512x512 grid.
  for (int e = t; e < kSN * kSN; e += kThreads) {
    const int sa = e / kSN, sb = e % kSN;
    const int a = A0 - 1 + sa, b = B0 - 1 + sb;
    float s = 0.0f;
    if (a >= 0 && a < kSH && b >= 0 && b < kSH) {
      const float* p = hu + (size_t)(2 * a) * kW + (size_t)(2 * b);
#pragma unroll
      for (int ch = 0; ch < kC; ++ch)
        s += (p[(size_t)ch * kHW] < kHoleP) ? 1.0f : 0.0f;
    }
    s_small[e] = s;
  }

  wait_async_zero();
  __syncthreads();

  // ---- 3) Initialize v, w over the 48x48 region ---------------------------
#pragma unroll
  for (int k = 0; k < kPxT; ++k) {
    const int idx = t + k * kThreads;
    const int r  = idx / kR, cc = idx % kR;
    const int gi = i0 - kHalo + r, gj = j0 - kHalo + cc;
    float v = 0.0f, w = 0.0f;
    if (gi >= 0 && gi < kH && gj >= 0 && gj < kW) {
      const bool inval = u_lds[idx] < kHoleP;
      // dilated union mask: clip(cross-conv(s_small))(gi>>1, gj>>1) > 0
      const int sa = (gi >> 1) - (A0 - 1);     // in [1, 24]
      const int sb = (gj >> 1) - (B0 - 1);
      const float d = s_small[sa * kSN + sb]
                    + s_small[(sa - 1) * kSN + sb]
                    + s_small[(sa + 1) * kSN + sb]
                    + s_small[sa * kSN + sb - 1]
                    + s_small[sa * kSN + sb + 1];
      bool srcpos = (d > 0.0f) && !inval;
      // corner fixup: src forced to 1 at the 4 image corners
      if ((gi == 0 || gi == kH - 1) && (gj == 0 || gj == kW - 1)) srcpos = true;
      if (srcpos) { w = 1.0f; v = inval ? 0.0f : x_lds[idx]; }
    }
    v_lds[idx] = v;
    w_lds[idx] = w;
  }
  __syncthreads();

  // ---- 4) 8 propagation iterations entirely in LDS ------------------------
  for (int it = 0; it < kNProp; ++it) {
    float cand[kPxT];
    int   flags = 0;
#pragma unroll
    for (int k = 0; k < kPxT; ++k) {
      const int idx = t + k * kThreads;
      const int r  = idx / kR, cc = idx % kR;
      const int gi = i0 - kHalo + r, gj = j0 - kHalo + cc;
      cand[k] = 0.0f;
      const bool inimg = (gi >= 0 && gi < kH && gj >= 0 && gj < kW);
      if (inimg && w_lds[idx] == 0.0f) {
        // full 3x3 cross (center contributes 0 since v==0 where w==0)
        float nv = v_lds[idx], nw = w_lds[idx];
        if (r > 0)       { nv += v_lds[idx - kR]; nw += w_lds[idx - kR]; }
        if (r < kR - 1)  { nv += v_lds[idx + kR]; nw += w_lds[idx + kR]; }
        if (cc > 0)      { nv += v_lds[idx - 1];  nw += w_lds[idx - 1];  }
        if (cc < kR - 1) { nv += v_lds[idx + 1];  nw += w_lds[idx + 1];  }
        if (nw > 0.0f) {
          cand[k] = nv / fmaxf(nw, 1e-6f);
          flags |= (1 << k);
        }
      }
    }
    __syncthreads();  // all reads done before any writes
#pragma unroll
    for (int k = 0; k < kPxT; ++k) {
      if (flags & (1 << k)) {
        const int idx = t + k * kThreads;
        v_lds[idx] = cand[k];
        w_lds[idx] = 1.0f;
      }
    }
    __syncthreads();
  }

  // ---- 5) Write the 32x32 tile (streaming / non-temporal) -----------------
#pragma unroll
  for (int k = 0; k < (kTS * kTS) / kThreads; ++k) {
    const int idxt = t + k * kThreads;
    const int ti = idxt / kTS, tj = idxt % kTS;
    const int gi = i0 + ti, gj = j0 + tj;
    const int idx = (ti + kHalo) * kR + (tj + kHalo);
    const bool inval = u_lds[idx] < kHoleP;
    const float res = inval ? v_lds[idx] : x_lds[idx];
    __builtin_nontemporal_store(res, &out[(size_t)c * kHW + (size_t)gi * kW + gj]);
  }
}

// ---------------------------------------------------------------------------
// Launch
// ---------------------------------------------------------------------------
extern "C" void kernel_launch(void* const* d_in, const int* in_sizes, int n_in,
                              void* d_out, int out_size, void* d_ws, size_t ws_size,
                              hipStream_t stream) {
  (void)in_sizes; (void)n_in; (void)out_size; (void)d_ws; (void)ws_size;
  const float* x  = (const float*)d_in[0];
  const float* hu = (const float*)d_in[1];
  float* out = (float*)d_out;
  dim3 grid(kW / kTS, kH / kTS, kC);   // 32 x 32 x 16 blocks
  hole_fill_kernel<<<grid, kThreads, 0, stream>>>(x, hu, out);
}